// _RandomForestPlusRMOE_9053791060044
// MI455X (gfx1250) — compile-verified
//
#include <hip/hip_runtime.h>
#include <hip/hip_bf16.h>

typedef __attribute__((ext_vector_type(2))) float v2f;
typedef __attribute__((ext_vector_type(4))) float v4f;
typedef __attribute__((ext_vector_type(8))) float v8f;

#define N_ROWS 32768
#define DIM    512
#define NE     128      // experts
#define NW     256      // 2*NE (scores | expert outputs)
#define NTOT   100000
#define NOOB   40000
#define ROWS_B 64       // rows per GEMM block
#define KG     (DIM/4)  // 128 k-groups of 4

// ---------------- kernel 1: clear table + accumulators ----------------
__global__ void k_clear(unsigned int* __restrict__ table, float* __restrict__ acc) {
    int i = blockIdx.x * blockDim.x + threadIdx.x;
    const int total = NTOT * 4;
    for (int j = i; j < total; j += gridDim.x * blockDim.x) table[j] = 0u;
    if (i < 2 * NE) acc[i] = 0.0f;
}

// ---------------- kernel 2: scatter oob bits ----------------
__global__ void k_scatter(const int* __restrict__ oob, unsigned int* __restrict__ table) {
    int i = blockIdx.x * 256 + threadIdx.x;
    if (i >= NE * NOOB) return;
    int e   = i / NOOB;
    int idx = oob[i];
    atomicOr(&table[idx * 4 + (e >> 5)], 1u << (e & 31));
}

// -------- kernel 3: pack W = [Wg | We^T] into B-fragment order --------
// Wp[(g*256 + n)*4 + kk] = W[4g+kk][n], so a lane's B operand (cols n,
// rows k+2h..k+2h+1) is one contiguous 8-byte load.
__global__ void k_packw(const float* __restrict__ Wg, const float* __restrict__ We,
                        float* __restrict__ Wp) {
    int i = blockIdx.x * 256 + threadIdx.x;
    if (i >= DIM * NW) return;
    int d = i >> 8, c = i & 255;                      // d = k row, c = output col
    float v = (c < NE) ? Wg[d * NE + c] : We[(c - NE) * DIM + d];
    Wp[((d >> 2) * NW + c) * 4 + (d & 3)] = v;
}

// ---------------- kernel 4: C[N x 256] = x[N x 512] @ W[512 x 256] ----------------
// 64 rows/block staged in LDS in A-fragment order with XOR swizzle; each of
// 8 waves owns a 32-wide N strip over all 64 rows: 8 accumulators, 8 WMMA/k-group.
__global__ __launch_bounds__(256) void k_gemm(const float* __restrict__ x,
                                              const float* __restrict__ Wp,
                                              float* __restrict__ C) {
    __shared__ float xs[ROWS_B * DIM];                // 128 KB, fragment-ordered
    const int t = threadIdx.x;
    const int rowBase = blockIdx.x * ROWS_B;

    // cooperative coalesced load: fragment f = row*128 + g (one float4 each)
    const v4f* xg = (const v4f*)(x + (size_t)rowBase * DIM);
    #pragma unroll
    for (int i = 0; i < 32; ++i) {
        int f   = t + 256 * i;                        // 0..8191
        int row = f >> 7;
        int g   = f & 127;
        int swz = (row & 0x30) | ((row ^ g) & 15);    // row ^ (g&15)
        *(v4f*)&xs[(g * ROWS_B + swz) * 4] = xg[f];
    }
    __syncthreads();

    const int wave = t >> 5, lane = t & 31;
    const int r = lane & 15, h = lane >> 4;
    const int n0 = wave * 32;                         // 8 waves x 32 cols

    v8f a00 = {}, a01 = {}, a10 = {}, a11 = {};
    v8f a20 = {}, a21 = {}, a30 = {}, a31 = {};

    const float* wb = Wp + (size_t)(n0 + r) * 4 + 2 * h;

    #pragma unroll 2
    for (int g = 0; g < KG; ++g) {
        const float* xa = &xs[g * (ROWS_B * 4) + ((r ^ (g & 15)) << 2) + 2 * h];
        v2f f0 = *(const v2f*)(xa);                   // M-tile 0 (rows 0-15)
        v2f f1 = *(const v2f*)(xa + 64);              // M-tile 1
        v2f f2 = *(const v2f*)(xa + 128);             // M-tile 2
        v2f f3 = *(const v2f*)(xa + 192);             // M-tile 3
        const float* wk = wb + (size_t)g * (NW * 4);
        v2f b0 = *(const v2f*)(wk);                   // N-tile 0
        v2f b1 = *(const v2f*)(wk + 64);              // N-tile 1
        a00 = __builtin_amdgcn_wmma_f32_16x16x4_f32(false, f0, false, b0, (short)0, a00, false, false);
        a01 = __builtin_amdgcn_wmma_f32_16x16x4_f32(false, f0, false, b1, (short)0, a01, false, false);
        a10 = __builtin_amdgcn_wmma_f32_16x16x4_f32(false, f1, false, b0, (short)0, a10, false, false);
        a11 = __builtin_amdgcn_wmma_f32_16x16x4_f32(false, f1, false, b1, (short)0, a11, false, false);
        a20 = __builtin_amdgcn_wmma_f32_16x16x4_f32(false, f2, false, b0, (short)0, a20, false, false);
        a21 = __builtin_amdgcn_wmma_f32_16x16x4_f32(false, f2, false, b1, (short)0, a21, false, false);
        a30 = __builtin_amdgcn_wmma_f32_16x16x4_f32(false, f3, false, b0, (short)0, a30, false, false);
        a31 = __builtin_amdgcn_wmma_f32_16x16x4_f32(false, f3, false, b1, (short)0, a31, false, false);
    }

    // C/D layout: VGPR v -> row v + 8h, col = lane%16
    float* crow = C + (size_t)rowBase * NW + n0 + r;
    #pragma unroll
    for (int v = 0; v < 8; ++v) {
        int ro0 = (v + 8 * h) * NW;
        crow[ro0]                 = a00[v];
        crow[ro0 + 16]            = a01[v];
        crow[ro0 + 16 * NW]       = a10[v];
        crow[ro0 + 16 * NW + 16]  = a11[v];
        crow[ro0 + 32 * NW]       = a20[v];
        crow[ro0 + 32 * NW + 16]  = a21[v];
        crow[ro0 + 48 * NW]       = a30[v];
        crow[ro0 + 48 * NW + 16]  = a31[v];
    }
}

// ---------------- kernel 5: masked softmax, gates, out, importance/load ----------------
__global__ __launch_bounds__(256) void k_epilogue(const float* __restrict__ C,
                                                  const int* __restrict__ index,
                                                  const unsigned int* __restrict__ table,
                                                  const float* __restrict__ bg,
                                                  const float* __restrict__ be,
                                                  float* __restrict__ outv,
                                                  float* __restrict__ gates,
                                                  float* __restrict__ impG,
                                                  float* __restrict__ loadG) {
    __shared__ float impS[NE], loadS[NE];
    const int t = threadIdx.x;
    if (t < NE) { impS[t] = 0.0f; loadS[t] = 0.0f; }
    __syncthreads();

    const int wave = t >> 5, lane = t & 31;
    for (int rr = 0; rr < 8; ++rr) {
        const int row = blockIdx.x * 64 + wave * 8 + rr;
        const int idx = index[row];
        const float* crow = C + (size_t)row * NW;

        float s[4], eo[4], ex[4];
        #pragma unroll
        for (int j = 0; j < 4; ++j) {
            int e = lane + 32 * j;                    // e/32 == j, bit == lane
            unsigned int w = table[idx * 4 + j];
            float sv = crow[e] + bg[e];
            s[j]  = ((w >> lane) & 1u) ? sv : 0.0f;   // scores * oob_mask
            eo[j] = crow[NE + e] + be[e];
        }
        float m = fmaxf(fmaxf(s[0], s[1]), fmaxf(s[2], s[3]));
        for (int off = 16; off >= 1; off >>= 1) m = fmaxf(m, __shfl_xor(m, off, 32));
        float sum = 0.0f;
        #pragma unroll
        for (int j = 0; j < 4; ++j) { ex[j] = __expf(s[j] - m); sum += ex[j]; }
        for (int off = 16; off >= 1; off >>= 1) sum += __shfl_xor(sum, off, 32);
        const float inv = 1.0f / sum;

        float o = 0.0f;
        #pragma unroll
        for (int j = 0; j < 4; ++j) {
            int e = lane + 32 * j;
            float g = ex[j] * inv;
            gates[(size_t)row * NE + e] = g;
            o += g * eo[j];
            atomicAdd(&impS[e], g);
            if (g > 0.0f) atomicAdd(&loadS[e], 1.0f);
        }
        for (int off = 16; off >= 1; off >>= 1) o += __shfl_xor(o, off, 32);
        if (lane == 0) outv[row] = o;
    }
    __syncthreads();
    if (t < NE) { atomicAdd(&impG[t], impS[t]); atomicAdd(&loadG[t], loadS[t]); }
}

// ---------------- kernel 6: loss = 0.01*(cv2(importance) + cv2(load)), ddof=1 ----------------
__global__ void k_loss(const float* __restrict__ impG, const float* __restrict__ loadG,
                       float* __restrict__ lossOut) {
    if (threadIdx.x != 0 || blockIdx.x != 0) return;
    float mi = 0.0f, ml = 0.0f;
    for (int e = 0; e < NE; ++e) { mi += impG[e]; ml += loadG[e]; }
    mi *= (1.0f / NE); ml *= (1.0f / NE);
    float vi = 0.0f, vl = 0.0f;
    for (int e = 0; e < NE; ++e) {
        float di = impG[e] - mi, dl = loadG[e] - ml;
        vi += di * di; vl += dl * dl;
    }
    vi *= (1.0f / (NE - 1)); vl *= (1.0f / (NE - 1));
    float cvi = vi / (mi * mi + 1e-10f);
    float cvl = vl / (ml * ml + 1e-10f);
    *lossOut = 0.01f * (cvi + cvl);
}

extern "C" void kernel_launch(void* const* d_in, const int* in_sizes, int n_in,
                              void* d_out, int out_size, void* d_ws, size_t ws_size,
                              hipStream_t stream) {
    const float* x   = (const float*)d_in[0];
    const float* Wg  = (const float*)d_in[1];
    const float* bg  = (const float*)d_in[2];
    const float* We  = (const float*)d_in[3];
    const float* be  = (const float*)d_in[4];
    const int*  index = (const int*)d_in[5];
    const int*  oob   = (const int*)d_in[6];

    char* ws = (char*)d_ws;
    unsigned int* table = (unsigned int*)ws;            // 1,600,000 B  (100000 x 128 bits)
    float* Wp  = (float*)(ws + 1600256);                // 524,288 B    (fragment-packed W)
    float* C   = (float*)(ws + 2124544);                // 33,554,432 B (32768 x 256)
    float* acc = (float*)(ws + 35678976);               // 1,024 B      (imp | load)
    float* impG  = acc;
    float* loadG = acc + NE;

    float* outv    = (float*)d_out;                     // [32768]
    float* lossOut = outv + N_ROWS;                     // [1]
    float* gates   = outv + N_ROWS + 1;                 // [32768 x 128]

    k_clear   <<<1564, 256, 0, stream>>>(table, acc);
    k_scatter <<<(NE * NOOB + 255) / 256, 256, 0, stream>>>(oob, table);
    k_packw   <<<(DIM * NW + 255) / 256, 256, 0, stream>>>(Wg, We, Wp);
    k_gemm    <<<N_ROWS / ROWS_B, 256, 0, stream>>>(x, Wp, C);
    k_epilogue<<<N_ROWS / 64, 256, 0, stream>>>(C, index, table, bg, be,
                                                outv, gates, impG, loadG);
    k_loss    <<<1, 32, 0, stream>>>(impG, loadG, lossOut);
}